// Theta_layer_4818953306928
// MI455X (gfx1250) — compile-verified
//
#include <hip/hip_runtime.h>

typedef __attribute__((ext_vector_type(2))) float v2f;
typedef __attribute__((ext_vector_type(8))) float v8f;

// s' = (1/(2*SIGMA)) * log2(e) = 2*pi*log2(e); exponents computed in base 2 so the
// hardware v_exp_f32 (which is exp2) needs no extra multiply.
#define S2      9.0647202836543876f
#define TWO_S2  18.129440567308775f

__device__ __forceinline__ float wexp2(float x) { return __builtin_amdgcn_exp2f(x); }

__global__ __launch_bounds__(256) void theta_wmma_kernel(
    const float* __restrict__ C,    // (B, N, 3)
    const float* __restrict__ dom,  // (K, 3)
    float* __restrict__ out,        // (B, K)
    int Bn, int Nn, int Kn)
{
  const int lane = threadIdx.x & 31;
  const int half = lane >> 4;   // 0: lanes 0-15, 1: lanes 16-31
  const int m    = lane & 15;

  const int wave   = (int)((blockIdx.x * blockDim.x + threadIdx.x) >> 5);
  const int ktiles = Kn >> 4;
  const int b      = wave / ktiles;
  const int k0     = (wave % ktiles) << 4;
  if (b >= Bn) return;  // grid sized exactly; wave-uniform, EXEC stays all-ones

  // ---- A matrix (per-wave constant). M = eval point (k0+m), Kdim = 4.
  // A[k] = [2s'*dx, 2s'*dy, 2s'*dz, -s'*||d_k||^2]
  // 32-bit 16x4 A layout: lanes 0-15 -> K=0 (v0), K=1 (v1); lanes 16-31 -> K=2 (v0), K=3 (v1)
  const float* dk = dom + (size_t)(k0 + m) * 3;
  const float dx = dk[0], dy = dk[1], dz = dk[2];
  const float phim = dx * dx + dy * dy + dz * dz;
  v2f Areg;
  Areg.x = half ? (TWO_S2 * dz) : (TWO_S2 * dx);
  Areg.y = half ? (-S2 * phim)  : (TWO_S2 * dy);

  v8f acc   = (v8f){0.f, 0.f, 0.f, 0.f, 0.f, 0.f, 0.f, 0.f};
  v8f zeroC = (v8f){0.f, 0.f, 0.f, 0.f, 0.f, 0.f, 0.f, 0.f};

  // Per-lane base pointer for atom n0+m; advanced by 16 atoms (192 B) per tile.
  const float* cn = C + (size_t)b * Nn * 3 + (size_t)m * 3;

  for (int n0 = 0; n0 < Nn; n0 += 16, cn += 48) {
    // Each lane loads all three components of its atom (one clause, offsets 0/4/8):
    const float cx = cn[0];
    const float cy = cn[1];
    const float cz = cn[2];

    // ---- B matrix tile. Kdim x N = 4 x 16, N = atom (n0+m).
    // B[.][n] = [Cx, Cy, Cz, 1]; low lanes: rows 0 (v0) & 1 (v1); high lanes: rows 2 (v0) & 3 (v1)
    v2f Breg;
    Breg.x = half ? cz : cx;
    Breg.y = half ? 1.0f : cy;

    // Per-atom factor computed locally (no cross-lane traffic):
    const float phib = fmaf(cx, cx, fmaf(cy, cy, cz * cz));
    const float E    = wexp2(-S2 * phib);

    // D[k][n] = 2s'*lam - s'*phim   (base-2 exponent minus the phib term)
    v8f D = __builtin_amdgcn_wmma_f32_16x16x4_f32(
        /*neg_a=*/false, Areg, /*neg_b=*/false, Breg,
        /*c_mod=*/(short)0, zeroC, /*reuse_a=*/false, /*reuse_b=*/false);

#pragma unroll
    for (int j = 0; j < 8; ++j)
      acc[j] = fmaf(wexp2(D[j]), E, acc[j]);
  }

  // Sum over n-columns: reduce across the 16 lanes of each half (once per wave).
#pragma unroll
  for (int off = 1; off < 16; off <<= 1) {
#pragma unroll
    for (int j = 0; j < 8; ++j)
      acc[j] += __shfl_xor(acc[j], off, 32);
  }

  // D-layout: VGPR j, lane half h -> row k = k0 + j + 8*h. Lanes m==0 hold full sums.
  if (m == 0) {
    float* op = out + (size_t)b * Kn + k0 + half * 8;
#pragma unroll
    for (int j = 0; j < 8; ++j) op[j] = acc[j];
  }
}

extern "C" void kernel_launch(void* const* d_in, const int* in_sizes, int n_in,
                              void* d_out, int out_size, void* d_ws, size_t ws_size,
                              hipStream_t stream) {
  const float* C   = (const float*)d_in[0];  // (32, N, 3)
  const float* dom = (const float*)d_in[1];  // (K, 3)
  float* out = (float*)d_out;                // (32, K)

  const int Bn = 32;
  const int Nn = in_sizes[0] / (Bn * 3);     // 4096
  const int Kn = in_sizes[1] / 3;            // 2048

  const int waves          = Bn * (Kn >> 4); // 4096 (one wave per (b, 16-k tile))
  const int wavesPerBlock  = 8;              // 256 threads
  const int blocks         = (waves + wavesPerBlock - 1) / wavesPerBlock;

  theta_wmma_kernel<<<blocks, wavesPerBlock * 32, 0, stream>>>(C, dom, out, Bn, Nn, Kn);
}